// MultiHeadEuclideanCodebook_67302137528710
// MI455X (gfx1250) — compile-verified
//
#include <hip/hip_runtime.h>

#define BB 8
#define TT 4096
#define DD 256
#define HH 8
#define KK 1024
#define DH 32

typedef __attribute__((ext_vector_type(2))) float v2f;
typedef __attribute__((ext_vector_type(8))) float v8f;

__global__ __launch_bounds__(256) void mhec_kernel(
    const float* __restrict__ x,       // [B,T,D]
    const float* __restrict__ embeds,  // [H,K,DH]
    float* __restrict__ out_q,         // [B,T,D]
    float* __restrict__ out_loss,      // [1]
    float* __restrict__ out_idx)       // [B,T,H] (written as float)
{
    __shared__ float s_e2[KK];

    const int h = blockIdx.y;
    const float* eh = embeds + (size_t)h * KK * DH;

    // Cooperative |e_k|^2 precompute into LDS (4 codes / thread).
    for (int c = threadIdx.x; c < KK; c += blockDim.x) {
        const float* ep = eh + c * DH;
        float s = 0.f;
        #pragma unroll
        for (int d = 0; d < DH; d += 4) {
            float4 v = *(const float4*)(ep + d);
            s += v.x * v.x + v.y * v.y + v.z * v.z + v.w * v.w;
        }
        s_e2[c] = s;
    }
    __syncthreads();

    if (blockIdx.x == 0 && blockIdx.y == 0 && threadIdx.x == 0)
        out_loss[0] = 0.0f;

    const int lane = threadIdx.x & 31;
    const int wid  = threadIdx.x >> 5;
    const int unit = blockIdx.x * (blockDim.x >> 5) + wid; // 0..1023
    const int tok0 = unit * 32;                            // 2 tiles of 16 tokens
    const int mrow = lane & 15;                            // A row / B col
    const int kh   = lane >> 4;                            // K-half select

    // A operand (16x4 f32 WMMA layout): lane holds row mrow, K-pair (2*kh, 2*kh+1)
    // for each of the 8 K-steps. Loaded once, reused across all 64 code tiles.
    v2f a[2][8];
    #pragma unroll
    for (int t = 0; t < 2; ++t) {
        const float* xp = x + (size_t)(tok0 + t * 16 + mrow) * DD + h * DH + kh * 2;
        #pragma unroll
        for (int kk = 0; kk < 8; ++kk)
            a[t][kk] = *(const v2f*)(xp + kk * 4);
    }

    float vmax[2][8];
    int   vidx[2][8];
    #pragma unroll
    for (int t = 0; t < 2; ++t)
        #pragma unroll
        for (int i = 0; i < 8; ++i) { vmax[t][i] = -3.4e38f; vidx[t][i] = 0; }

    for (int ct = 0; ct < KK / 16; ++ct) {
        const int code = ct * 16 + mrow;
        // B operand (4x16): lane holds column `mrow`, same K-pair striping as A.
        const float* ep = eh + (size_t)code * DH + kh * 2;
        v2f bm[8];
        #pragma unroll
        for (int kk = 0; kk < 8; ++kk)
            bm[kk] = *(const v2f*)(ep + kk * 4);
        const float e2l = s_e2[code];

        #pragma unroll
        for (int t = 0; t < 2; ++t) {
            v8f c = {};
            #pragma unroll
            for (int kk = 0; kk < 8; ++kk)
                c = __builtin_amdgcn_wmma_f32_16x16x4_f32(
                        false, a[t][kk], false, bm[kk],
                        (short)0, c, false, false);
            // score = 2*dot - |e|^2 (argmax-equivalent to reference dist)
            #pragma unroll
            for (int i = 0; i < 8; ++i) {
                float s = fmaf(2.0f, c[i], -e2l);
                if (s > vmax[t][i]) { vmax[t][i] = s; vidx[t][i] = code; }
            }
        }
    }

    // Cross-lane argmax within each 16-lane group (codes live across lanes).
    #pragma unroll
    for (int t = 0; t < 2; ++t)
        #pragma unroll
        for (int i = 0; i < 8; ++i) {
            float m = vmax[t][i]; int id = vidx[t][i];
            #pragma unroll
            for (int off = 8; off >= 1; off >>= 1) {
                float om = __shfl_xor(m, off, 32);
                int   oi = __shfl_xor(id, off, 32);
                if (om > m || (om == m && oi < id)) { m = om; id = oi; }
            }
            vmax[t][i] = m; vidx[t][i] = id;
        }

    // Gather winning code rows (32 lanes cover DH=32) + write indices.
    #pragma unroll
    for (int t = 0; t < 2; ++t) {
        #pragma unroll
        for (int r = 0; r < 16; ++r) {
            const int id  = __shfl(vidx[t][r & 7], (r & 8) ? 16 : 0, 32);
            const int tok = tok0 + t * 16 + r;
            out_q[(size_t)tok * DD + h * DH + lane] = eh[(size_t)id * DH + lane];
            if (lane == 0)
                out_idx[(size_t)tok * HH + h] = (float)id;
        }
    }
}

extern "C" void kernel_launch(void* const* d_in, const int* in_sizes, int n_in,
                              void* d_out, int out_size, void* d_ws, size_t ws_size,
                              hipStream_t stream) {
    const float* x      = (const float*)d_in[0];
    // d_in[1] = x_len (unused by eval-mode reference)
    const float* embeds = (const float*)d_in[2];

    float* out      = (float*)d_out;
    float* out_q    = out;                          // B*T*D
    float* out_loss = out + (size_t)BB * TT * DD;   // 1
    float* out_idx  = out_loss + 1;                 // B*T*H

    dim3 grid(128, HH);   // 128 blocks * 8 waves * 32 tokens = 32768 tokens
    mhec_kernel<<<grid, 256, 0, stream>>>(x, embeds, out_q, out_loss, out_idx);
}